// GATEncoder_80719615361069
// MI455X (gfx1250) — compile-verified
//
#include <hip/hip_runtime.h>
#include <math.h>

typedef __attribute__((ext_vector_type(2))) float v2f;
typedef __attribute__((ext_vector_type(8))) float v8f;

#define NEG_SLOPE 0.2f

// ---- ordered-uint mapping so atomicMax(uint) == float max (handles negatives) ----
__device__ __forceinline__ unsigned f2ord(float f) {
  unsigned u = __float_as_uint(f);
  return (u & 0x80000000u) ? ~u : (u | 0x80000000u);
}
__device__ __forceinline__ float ord2f(unsigned v) {
  unsigned u = (v & 0x80000000u) ? (v ^ 0x80000000u) : ~v;
  return __uint_as_float(u);
}

__device__ __forceinline__ void edge_src_dst(const int* __restrict__ ei, int E,
                                             int e, int& s, int& d) {
  if (e < E) { s = ei[e]; d = ei[E + e]; }
  else       { s = e - E; d = e - E; }   // self loop appended at end
}

// ---------------------------------------------------------------------------
// GEMM: C[M,N] = A[M,256] @ B[256,N], fp32.  K fixed at 256 (true for both
// layers).  Block = 256 threads = 8 waves computing a 16x128 strip:
//   - 16x256 A panel staged into LDS once via global_load_async_to_lds_b128
//     (ASYNCcnt -> s_wait_asynccnt), rows padded to 260 floats (bank spread)
//   - each wave owns one 16-col tile, V_WMMA_F32_16X16X4_F32 over K.
// Grid must be exactly (M/16) * (N/128).
// ---------------------------------------------------------------------------
__global__ __launch_bounds__(256)
void gemm_f32_wmma_lds(const float* __restrict__ A, const float* __restrict__ B,
                       float* __restrict__ C, int M, int N) {
  constexpr int K   = 256;
  constexpr int LDA = 260;                    // padded LDS row stride (floats)
  __shared__ float Atile[16 * LDA];

  const int lane = threadIdx.x & 31;
  const int wave = threadIdx.x >> 5;
  const int nBlk = N >> 7;                    // # of 128-col strips
  const int tm   = blockIdx.x / nBlk;
  const int nb   = blockIdx.x - tm * nBlk;

  // ---- async-stage the 16x256 A panel into LDS (64 B per thread) ----
  {
    const int row = threadIdx.x >> 4;         // 0..15
    const int seg = (threadIdx.x & 15) << 4;  // float offset 0,16,...,240
    const float* g = A + (size_t)(tm * 16 + row) * K + seg;
    unsigned l = (unsigned)(size_t)(&Atile[row * LDA + seg]);
    asm volatile(
        "global_load_async_to_lds_b128 %0, %1, off\n\t"
        "global_load_async_to_lds_b128 %0, %1, off offset:16\n\t"
        "global_load_async_to_lds_b128 %0, %1, off offset:32\n\t"
        "global_load_async_to_lds_b128 %0, %1, off offset:48"
        :: "v"(l), "v"(g) : "memory");
    asm volatile("s_wait_asynccnt 0" ::: "memory");
  }
  __syncthreads();

  const int r16  = lane & 15;                 // row (A frag) / col (B frag)
  const int koff = (lane >> 4) << 1;          // lanes 0-15 -> K0,K1 ; 16-31 -> K2,K3
  const int col0 = (nb * 8 + wave) * 16;

  const float* Ap = &Atile[r16 * LDA + koff];           // LDS
  const float* Bp = B + (size_t)koff * N + col0 + r16;  // global (L2-resident)

  v8f acc = {};
  for (int k = 0; k < K; k += 4) {
    v2f a, b;
    a.x = Ap[k];
    a.y = Ap[k + 1];
    b.x = Bp[(size_t)k * N];
    b.y = Bp[(size_t)(k + 1) * N];
    acc = __builtin_amdgcn_wmma_f32_16x16x4_f32(false, a, false, b,
                                                (short)0, acc, false, false);
  }

  // C/D layout: VGPR r, lanes 0-15 -> M=r, lanes 16-31 -> M=r+8 ; N = lane&15
  float* Cp = C + (size_t)(tm * 16 + ((lane >> 4) << 3)) * N + col0 + r16;
#pragma unroll
  for (int r = 0; r < 8; ++r)
    Cp[(size_t)r * N] = acc[r];
}

// ---------------------------------------------------------------------------
// alpha_src / alpha_dst: per (node, head) dot products over C channels.
// One wave per (n,h).
// ---------------------------------------------------------------------------
__global__ __launch_bounds__(256)
void alpha_kernel(const float* __restrict__ Hm, const float* __restrict__ a_src,
                  const float* __restrict__ a_dst, float* __restrict__ as,
                  float* __restrict__ ad, int N, int Hh, int C) {
  const int gw   = (int)((blockIdx.x * (unsigned)blockDim.x + threadIdx.x) >> 5);
  const int lane = threadIdx.x & 31;
  if (gw >= N * Hh) return;
  const int n = gw / Hh, h = gw - n * Hh;
  const float* hp  = Hm + (size_t)n * Hh * C + (size_t)h * C;
  const float* asp = a_src + (size_t)h * C;
  const float* adp = a_dst + (size_t)h * C;
  float ss = 0.f, sd = 0.f;
  for (int c = lane; c < C; c += 32) {
    const float v = hp[c];
    ss += v * asp[c];
    sd += v * adp[c];
  }
#pragma unroll
  for (int o = 16; o > 0; o >>= 1) {
    ss += __shfl_down(ss, o, 32);
    sd += __shfl_down(sd, o, 32);
  }
  if (lane == 0) { as[gw] = ss; ad[gw] = sd; }
}

// ---------------------------------------------------------------------------
// Segment max of leaky-relu edge logits (ordered-uint atomicMax).
// One thread per (edge, head).
// ---------------------------------------------------------------------------
__global__ __launch_bounds__(256)
void edge_max_kernel(const int* __restrict__ ei, int E, int Etot, int Hh,
                     const float* __restrict__ as, const float* __restrict__ ad,
                     unsigned* __restrict__ m) {
  const long long idx = (long long)blockIdx.x * blockDim.x + threadIdx.x;
  if (idx >= (long long)Etot * Hh) return;
  const int e = (int)(idx / Hh), h = (int)(idx - (long long)e * Hh);
  int s, d; edge_src_dst(ei, E, e, s, d);
  float v = as[s * Hh + h] + ad[d * Hh + h];
  v = (v > 0.f) ? v : v * NEG_SLOPE;
  atomicMax(&m[d * Hh + h], f2ord(v));
}

// ---------------------------------------------------------------------------
// Segment sum of exp(e - max).  One thread per (edge, head).
// ---------------------------------------------------------------------------
__global__ __launch_bounds__(256)
void edge_sum_kernel(const int* __restrict__ ei, int E, int Etot, int Hh,
                     const float* __restrict__ as, const float* __restrict__ ad,
                     const unsigned* __restrict__ m, float* __restrict__ ssum) {
  const long long idx = (long long)blockIdx.x * blockDim.x + threadIdx.x;
  if (idx >= (long long)Etot * Hh) return;
  const int e = (int)(idx / Hh), h = (int)(idx - (long long)e * Hh);
  int s, d; edge_src_dst(ei, E, e, s, d);
  float v = as[s * Hh + h] + ad[d * Hh + h];
  v = (v > 0.f) ? v : v * NEG_SLOPE;
  const float ex = expf(v - ord2f(m[d * Hh + h]));
  atomicAdd(&ssum[d * Hh + h], ex);
}

// ---------------------------------------------------------------------------
// Message aggregation: out[dst,h,:] += h[src,h,:] * softmax_weight.
// One wave per EDGE covering the full contiguous H*C row; per-head weights
// computed once by lanes 0..H-1 and broadcast with shfl.
// ---------------------------------------------------------------------------
__global__ __launch_bounds__(256)
void aggregate_kernel(const int* __restrict__ ei, int E, int Etot, int Hh, int C,
                      const float* __restrict__ as, const float* __restrict__ ad,
                      const unsigned* __restrict__ m, const float* __restrict__ ssum,
                      const float* __restrict__ Hm, float* __restrict__ out) {
  const long long gw = ((long long)blockIdx.x * blockDim.x + threadIdx.x) >> 5;
  const int lane = threadIdx.x & 31;
  if (gw >= (long long)Etot) return;
  const int e = (int)gw;
  int s, d; edge_src_dst(ei, E, e, s, d);

  // lane computes the softmax weight for head (lane mod Hh); Hh is 1 or 4
  const int hh = lane & (Hh - 1);
  float v = as[s * Hh + hh] + ad[d * Hh + hh];
  v = (v > 0.f) ? v : v * NEG_SLOPE;
  const float w = expf(v - ord2f(m[d * Hh + hh])) / (ssum[d * Hh + hh] + 1e-16f);

  const int HC     = Hh * C;
  const int cshift = 31 - __clz(C);          // log2(C); C is 64 or 128
  const float* hp  = Hm + (size_t)s * HC;
  float* op        = out + (size_t)d * HC;
  for (int c = lane; c < HC; c += 32) {
    const float wh = __shfl(w, c >> cshift, 32);
    atomicAdd(&op[c], hp[c] * wh);
  }
}

// ---------------------------------------------------------------------------
// X[n, j] = (relu?) (X[n, j] + b[j])
// ---------------------------------------------------------------------------
__global__ __launch_bounds__(256)
void bias_act_kernel(float* __restrict__ X, const float* __restrict__ b,
                     long long total, int cols, int do_relu) {
  const long long i = (long long)blockIdx.x * blockDim.x + threadIdx.x;
  if (i >= total) return;
  float v = X[i] + b[(int)(i % cols)];
  if (do_relu) v = fmaxf(v, 0.f);
  X[i] = v;
}

// ---------------------------------------------------------------------------
extern "C" void kernel_launch(void* const* d_in, const int* in_sizes, int n_in,
                              void* d_out, int out_size, void* d_ws, size_t ws_size,
                              hipStream_t stream) {
  const float* x   = (const float*)d_in[0];
  const int*   ei  = (const int*)d_in[1];
  const float* W1  = (const float*)d_in[2];
  const float* as1w= (const float*)d_in[3];
  const float* ad1w= (const float*)d_in[4];
  const float* b1  = (const float*)d_in[5];
  const float* W2  = (const float*)d_in[6];
  const float* as2w= (const float*)d_in[7];
  const float* ad2w= (const float*)d_in[8];
  const float* b2  = (const float*)d_in[9];

  const int F_in = 256, H1 = 4, C1 = 64, C2 = 128;
  const int N    = in_sizes[0] / F_in;     // 50000
  const int E    = in_sizes[1] / 2;        // 800000
  const int Etot = E + N;                  // + self loops

  // ---- workspace carve-up (floats) ----
  float* w = (float*)d_ws;
  float*    H1m = w;                 w += (size_t)N * H1 * C1;   // [N,256]
  float*    H1a = w;                 w += (size_t)N * H1 * C1;   // [N,256]
  float*    as1 = w;                 w += (size_t)N * H1;
  float*    ad1 = w;                 w += (size_t)N * H1;
  unsigned* m1  = (unsigned*)w;      w += (size_t)N * H1;
  float*    s1  = w;                 w += (size_t)N * H1;
  float*    H2m = w;                 w += (size_t)N * C2;        // [N,128]
  float*    as2 = w;                 w += (size_t)N;
  float*    ad2 = w;                 w += (size_t)N;
  unsigned* m2  = (unsigned*)w;      w += (size_t)N;
  float*    s2  = w;                 w += (size_t)N;
  (void)ws_size; (void)n_in; (void)out_size;

  float* out = (float*)d_out;

  // ---- zero / init accumulators (capture-safe memsets) ----
  hipMemsetAsync(H1a, 0, (size_t)N * H1 * C1 * sizeof(float), stream);
  hipMemsetAsync(s1,  0, (size_t)N * H1 * sizeof(float), stream);
  hipMemsetAsync(m1,  0, (size_t)N * H1 * sizeof(unsigned), stream);  // 0 == ordered max identity
  hipMemsetAsync(s2,  0, (size_t)N * sizeof(float), stream);
  hipMemsetAsync(m2,  0, (size_t)N * sizeof(unsigned), stream);
  hipMemsetAsync(out, 0, (size_t)N * C2 * sizeof(float), stream);

  const int TB = 256;

  // ================= Layer 1 =================
  {
    gemm_f32_wmma_lds<<<(N / 16) * ((H1 * C1) / 128), TB, 0, stream>>>(x, W1, H1m, N, H1 * C1);

    const int waves = N * H1;
    alpha_kernel<<<(waves + 7) / 8, TB, 0, stream>>>(H1m, as1w, ad1w, as1, ad1, N, H1, C1);

    const long long et = (long long)Etot * H1;
    edge_max_kernel<<<(int)((et + TB - 1) / TB), TB, 0, stream>>>(ei, E, Etot, H1, as1, ad1, m1);
    edge_sum_kernel<<<(int)((et + TB - 1) / TB), TB, 0, stream>>>(ei, E, Etot, H1, as1, ad1, m1, s1);
    aggregate_kernel<<<(Etot + 7) / 8, TB, 0, stream>>>(ei, E, Etot, H1, C1,
                                                        as1, ad1, m1, s1, H1m, H1a);

    const long long tot = (long long)N * H1 * C1;
    bias_act_kernel<<<(int)((tot + TB - 1) / TB), TB, 0, stream>>>(H1a, b1, tot, H1 * C1, 1);
  }

  // ================= Layer 2 =================
  {
    gemm_f32_wmma_lds<<<(N / 16) * (C2 / 128), TB, 0, stream>>>(H1a, W2, H2m, N, C2);

    alpha_kernel<<<(N + 7) / 8, TB, 0, stream>>>(H2m, as2w, ad2w, as2, ad2, N, 1, C2);

    const long long et = (long long)Etot;
    edge_max_kernel<<<(int)((et + TB - 1) / TB), TB, 0, stream>>>(ei, E, Etot, 1, as2, ad2, m2);
    edge_sum_kernel<<<(int)((et + TB - 1) / TB), TB, 0, stream>>>(ei, E, Etot, 1, as2, ad2, m2, s2);
    aggregate_kernel<<<(Etot + 7) / 8, TB, 0, stream>>>(ei, E, Etot, 1, C2,
                                                        as2, ad2, m2, s2, H2m, out);

    const long long tot = (long long)N * C2;
    bias_act_kernel<<<(int)((tot + TB - 1) / TB), TB, 0, stream>>>(out, b2, tot, C2, 0);
  }
}